// GRUDec_46926812676488
// MI455X (gfx1250) — compile-verified
//
#include <hip/hip_runtime.h>
#include <hip/hip_bf16.h>
#include <math.h>

// Problem constants (from reference)
#define B_   16
#define L_   128
#define H_   1024
#define V_   50257
#define T_   (L_ - 1)          // 127 identical output steps
#define VPAD 50272             // V rounded up to multiple of 16 (3142 tiles)
#define NTILES ((VPAD) / 16)   // 3142

typedef __attribute__((ext_vector_type(2))) float v2f;
typedef __attribute__((ext_vector_type(8))) float v8f;

// ---------------------------------------------------------------------------
// Kernel 1: x0 = relu(emb[input[:,0]]) ;  h0 = einsum("blh,l->bh") + bridge_b
// grid: 64 blocks x 256 threads covering B*H elements
// ---------------------------------------------------------------------------
__global__ void k_embed_bridge(const int* __restrict__ input,
                               const float* __restrict__ hidden,
                               const float* __restrict__ emb,
                               const float* __restrict__ bridge_w,
                               const float* __restrict__ bridge_b,
                               float* __restrict__ x0,
                               float* __restrict__ h0) {
    int idx = blockIdx.x * blockDim.x + threadIdx.x;   // [0, B*H)
    int b = idx >> 10;         // / H_
    int h = idx & (H_ - 1);    // % H_

    // embedding gather + relu
    int tok = input[b * L_ + 0];
    float xv = emb[(size_t)tok * H_ + h];
    x0[idx] = xv > 0.0f ? xv : 0.0f;

    // bridge: weighted sum over L
    float acc = bridge_b[0];
    const float* hp = hidden + ((size_t)b * L_) * H_ + h;
#pragma unroll 4
    for (int l = 0; l < L_; ++l) {
        acc = fmaf(hp[(size_t)l * H_], bridge_w[l], acc);
    }
    h0[idx] = acc;
}

// ---------------------------------------------------------------------------
// Kernel 2: single GRU cell step -> h1[B,H]
// grid: 64 blocks x 256 threads; block (blockIdx.x>>2) owns batch b,
// thread owns one gate-column j.
// ---------------------------------------------------------------------------
__global__ void k_gru_cell(const float* __restrict__ x0,
                           const float* __restrict__ h0,
                           const float* __restrict__ w_ih,
                           const float* __restrict__ w_hh,
                           const float* __restrict__ b_ih,
                           const float* __restrict__ b_hh,
                           float* __restrict__ h1) {
    int b = blockIdx.x >> 2;
    int j = ((blockIdx.x & 3) << 8) + threadIdx.x;     // [0, H)

    const float* xr = x0 + b * H_;
    const float* hr = h0 + b * H_;
    const float* wi_r = w_ih + (size_t)j * H_;
    const float* wi_z = w_ih + (size_t)(H_ + j) * H_;
    const float* wi_n = w_ih + (size_t)(2 * H_ + j) * H_;
    const float* wh_r = w_hh + (size_t)j * H_;
    const float* wh_z = w_hh + (size_t)(H_ + j) * H_;
    const float* wh_n = w_hh + (size_t)(2 * H_ + j) * H_;

    float ir = b_ih[j],        iz = b_ih[H_ + j],  inn = b_ih[2 * H_ + j];
    float hrv = b_hh[j],       hz = b_hh[H_ + j],  hn  = b_hh[2 * H_ + j];

#pragma unroll 4
    for (int k = 0; k < H_; ++k) {
        float x = xr[k];
        float h = hr[k];
        ir  = fmaf(x, wi_r[k], ir);
        iz  = fmaf(x, wi_z[k], iz);
        inn = fmaf(x, wi_n[k], inn);
        hrv = fmaf(h, wh_r[k], hrv);
        hz  = fmaf(h, wh_z[k], hz);
        hn  = fmaf(h, wh_n[k], hn);
    }

    float r = 1.0f / (1.0f + __expf(-(ir + hrv)));
    float z = 1.0f / (1.0f + __expf(-(iz + hz)));
    float n = tanhf(inn + r * hn);
    float h0v = h0[b * H_ + j];
    h1[b * H_ + j] = (1.0f - z) * n + z * h0v;
}

// ---------------------------------------------------------------------------
// Kernel 3: logits[16, VPAD] = h1[16,1024] @ proj_w.T + proj_b  via
// V_WMMA_F32_16X16X4_F32. One wave per 16x16 (batch x vocab) tile.
//
// f32 A-matrix 16x4 layout (ISA 7.12.2): lanes 0-15 hold M=0..15 with K={0,1},
// lanes 16-31 hold M=0..15 with K={2,3}  -> lane loads float2 at k + 2*half.
// B-matrix 4x16 is the mirrored layout   -> same float2 offset, row = v0+ln.
// D layout: VGPR i = row (i + 8*half), col = lane&15.
// ---------------------------------------------------------------------------
__global__ void k_proj_wmma(const float* __restrict__ h1,
                            const float* __restrict__ proj_w,
                            const float* __restrict__ proj_b,
                            float* __restrict__ logits) {
    int wave = (blockIdx.x << 3) + (threadIdx.x >> 5);  // 8 waves / block
    if (wave >= NTILES) return;                          // wave-uniform exit

    int lane = threadIdx.x & 31;
    int half = lane >> 4;       // 0: K lo pair, 1: K hi pair
    int ln   = lane & 15;

    int v0 = wave << 4;
    int v  = v0 + ln;
    int vrow = v < V_ ? v : (V_ - 1);   // clamp tail tile (pad cols discarded)

    const float* arow = h1 + (size_t)ln * H_;            // A: batch row = ln
    const float* brow = proj_w + (size_t)vrow * H_;      // B: vocab row

    v8f acc = {};
    for (int k = 0; k < H_; k += 16) {
#pragma unroll
        for (int u = 0; u < 4; ++u) {
            int kk = k + (u << 2) + (half << 1);
            v2f a = *reinterpret_cast<const v2f*>(arow + kk);
            v2f bm = *reinterpret_cast<const v2f*>(brow + kk);
            acc = __builtin_amdgcn_wmma_f32_16x16x4_f32(
                /*neg_a=*/false, a, /*neg_b=*/false, bm,
                /*c_mod=*/(short)0, acc, /*reuse_a=*/false, /*reuse_b=*/false);
        }
    }

    float bias = proj_b[vrow];
#pragma unroll
    for (int i = 0; i < 8; ++i) {
        int brow_m = i + (half << 3);                    // batch row of D
        logits[(size_t)brow_m * VPAD + v0 + ln] = acc[i] + bias;
    }
}

// ---------------------------------------------------------------------------
// Kernel 4: per-batch log-softmax stats: stats[b] = max, stats[16+b] = log(sum)
// 16 blocks (one per batch) x 256 threads, two-pass with LDS reduction.
// ---------------------------------------------------------------------------
__global__ void k_row_stats(const float* __restrict__ logits,
                            float* __restrict__ stats) {
    __shared__ float red[256];
    int b = blockIdx.x;
    int tid = threadIdx.x;
    const float* row = logits + (size_t)b * VPAD;

    float m = -INFINITY;
    for (int v = tid; v < V_; v += 256) m = fmaxf(m, row[v]);
    red[tid] = m;
    __syncthreads();
    for (int s = 128; s > 0; s >>= 1) {
        if (tid < s) red[tid] = fmaxf(red[tid], red[tid + s]);
        __syncthreads();
    }
    m = red[0];
    __syncthreads();

    float sum = 0.0f;
    for (int v = tid; v < V_; v += 256) sum += __expf(row[v] - m);
    red[tid] = sum;
    __syncthreads();
    for (int s = 128; s > 0; s >>= 1) {
        if (tid < s) red[tid] += red[tid + s];
        __syncthreads();
    }
    if (tid == 0) {
        stats[b] = m;
        stats[16 + b] = __logf(red[0]);
    }
}

// ---------------------------------------------------------------------------
// Kernel 5: broadcast write  out[b, t, v] = logits[b,v] - max_b - logsum_b
// grid: (ceil(V/256), B*T). This is the 408 MB store stream (the true floor).
// ---------------------------------------------------------------------------
__global__ void k_logsoftmax_bcast(const float* __restrict__ logits,
                                   const float* __restrict__ stats,
                                   float* __restrict__ out) {
    int v = blockIdx.x * blockDim.x + threadIdx.x;
    if (v >= V_) return;
    int bt = blockIdx.y;           // [0, B*T)
    int b = bt / T_;
    float val = logits[(size_t)b * VPAD + v] - stats[b] - stats[16 + b];
    out[(size_t)bt * V_ + v] = val;
}

// ---------------------------------------------------------------------------
extern "C" void kernel_launch(void* const* d_in, const int* in_sizes, int n_in,
                              void* d_out, int out_size, void* d_ws, size_t ws_size,
                              hipStream_t stream) {
    const int*   input    = (const int*)d_in[0];
    const float* hidden   = (const float*)d_in[1];
    const float* emb      = (const float*)d_in[2];
    const float* bridge_w = (const float*)d_in[3];
    const float* bridge_b = (const float*)d_in[4];
    const float* w_ih     = (const float*)d_in[5];
    const float* w_hh     = (const float*)d_in[6];
    const float* b_ih     = (const float*)d_in[7];
    const float* b_hh     = (const float*)d_in[8];
    const float* proj_w   = (const float*)d_in[9];
    const float* proj_b   = (const float*)d_in[10];
    float* out = (float*)d_out;

    // workspace layout (floats): x0 | h0 | h1 | stats | logits(16 x VPAD)
    float* ws     = (float*)d_ws;
    float* x0     = ws;                    // 16*1024
    float* h0     = ws + 16384;            // 16*1024
    float* h1     = ws + 32768;            // 16*1024
    float* stats  = ws + 49152;            // 32
    float* logits = ws + 49184;            // 16*50272  (~3.2 MB)

    k_embed_bridge<<<64, 256, 0, stream>>>(input, hidden, emb, bridge_w,
                                           bridge_b, x0, h0);
    k_gru_cell<<<64, 256, 0, stream>>>(x0, h0, w_ih, w_hh, b_ih, b_hh, h1);

    int blocksC = (NTILES + 7) / 8;        // 8 waves (wave32) per 256-thr block
    k_proj_wmma<<<blocksC, 256, 0, stream>>>(h1, proj_w, proj_b, logits);

    k_row_stats<<<B_, 256, 0, stream>>>(logits, stats);

    dim3 gB((V_ + 255) / 256, B_ * T_);
    k_logsoftmax_bcast<<<gB, 256, 0, stream>>>(logits, stats, out);
}